// Encoder_17291538333915
// MI455X (gfx1250) — compile-verified
//
#include <hip/hip_runtime.h>
#include <hip/hip_bf16.h>

#define NPART 16384
#define MAXE  (32 * NPART)   // 524288
#define NTILES (NPART / 16)  // 1024

typedef __attribute__((ext_vector_type(2))) float v2f;
typedef __attribute__((ext_vector_type(8))) float v8f;

__device__ __forceinline__ v8f wmma4(v2f a, v2f b, v8f c) {
  // D = A(16x4 f32) * B(4x16 f32) + C(16x16 f32)
  return __builtin_amdgcn_wmma_f32_16x16x4_f32(false, a, false, b, (short)0, c,
                                               false, false);
}

// ---------------------------------------------------------------------------
// Pack last positions (N,2) into contiguous ws buffer for coalesced tile loads
// ---------------------------------------------------------------------------
__global__ void __launch_bounds__(256)
prep_last_kernel(const float* __restrict__ pos, float2* __restrict__ last) {
  int g = blockIdx.x * 256 + threadIdx.x;
  last[g] = make_float2(pos[g * 12 + 10], pos[g * 12 + 11]);
}

// ---------------------------------------------------------------------------
// MLP (16->32->64->128) + LayerNorm, WMMA f32 16x16x4, one wave per 16 rows
// ---------------------------------------------------------------------------
__global__ void __launch_bounds__(128)
mlp_kernel(const float* __restrict__ pos, const float* __restrict__ bounds,
           const float* __restrict__ W1, const float* __restrict__ B1,
           const float* __restrict__ W2, const float* __restrict__ B2,
           const float* __restrict__ W3, const float* __restrict__ B3,
           const float* __restrict__ gamma, const float* __restrict__ beta,
           float* __restrict__ out_x) {
  __shared__ float h1s[4][16 * 32];
  __shared__ float h2s[4][16 * 64];
  const int lane = threadIdx.x & 31;
  const int w = threadIdx.x >> 5;
  const int m = lane & 15;
  const int hi = lane >> 4;
  const int rowTile = blockIdx.x * 4 + w;
  const int row = rowTile * 16 + m;

  // ---- per-row 16 features as NAMED SCALARS (no alloca -> no dynamic GEP
  //      -> no cndmask cascades). Lanes l and l+16 duplicate row m. ----
  const float4* prow = reinterpret_cast<const float4*>(pos + row * 12);
  const float4 p0 = prow[0], p1 = prow[1], p2 = prow[2];
  const float f0 = p0.z - p0.x, f1 = p0.w - p0.y;
  const float f2 = p1.x - p0.z, f3 = p1.y - p0.w;
  const float f4 = p1.z - p1.x, f5 = p1.w - p1.y;
  const float f6 = p2.x - p1.z, f7 = p2.y - p1.w;
  const float f8 = p2.z - p2.x, f9 = p2.w - p2.y;
  const float lx = p2.z, ly = p2.w;
  const float f10 = lx, f11 = ly;
  const float bx0 = bounds[0], bx1 = bounds[1], by0 = bounds[2], by1 = bounds[3];
  const float f12 = fminf(fmaxf(lx - bx0, -1.f), 1.f);
  const float f13 = fminf(fmaxf(lx - bx1, -1.f), 1.f);
  const float f14 = fminf(fmaxf(ly - by0, -1.f), 1.f);
  const float f15 = fminf(fmaxf(ly - by1, -1.f), 1.f);

  // A operands for layer 1: lanes<16 hold K=4kt..4kt+1, lanes>=16 K=4kt+2..+3.
  // Ternaries on scalars -> exactly one v_cndmask each.
  const bool H = (hi != 0);
  v2f a1[4];
  a1[0].x = H ? f2 : f0;
  a1[0].y = H ? f3 : f1;
  a1[1].x = H ? f6 : f4;
  a1[1].y = H ? f7 : f5;
  a1[2].x = H ? f10 : f8;
  a1[2].y = H ? f11 : f9;
  a1[3].x = H ? f14 : f12;
  a1[3].y = H ? f15 : f13;

  // ---- layer 1: (16x16) @ (16x32), K=16 -> 4 wmma steps, 2 col tiles ----
#pragma unroll
  for (int nt = 0; nt < 2; nt++) {
    const int n = nt * 16 + m;
    v8f acc = {};
#pragma unroll
    for (int kt = 0; kt < 4; kt++) {
      const int k0 = kt * 4 + 2 * hi;
      v2f b;
      b.x = W1[k0 * 32 + n];
      b.y = W1[(k0 + 1) * 32 + n];
      acc = wmma4(a1[kt], b, acc);
    }
    const float bias = B1[n];
#pragma unroll
    for (int v = 0; v < 8; v++) {
      float h = acc[v] + bias;
      h = h > 0.f ? h : 0.f;
      h1s[w][(v + 8 * hi) * 32 + n] = h;  // C-layout -> row-major LDS
    }
  }
  asm volatile("s_wait_dscnt 0x0" ::: "memory");  // per-wave LDS transpose sync

  // ---- layer 2: (16x32) @ (32x64), K=32 -> 8 wmma steps, 4 col tiles ----
#pragma unroll
  for (int nt = 0; nt < 4; nt++) {
    const int n = nt * 16 + m;
    v8f acc = {};
#pragma unroll
    for (int kt = 0; kt < 8; kt++) {
      const int k0 = kt * 4 + 2 * hi;
      v2f a;
      a.x = h1s[w][m * 32 + k0];
      a.y = h1s[w][m * 32 + k0 + 1];
      v2f b;
      b.x = W2[k0 * 64 + n];
      b.y = W2[(k0 + 1) * 64 + n];
      acc = wmma4(a, b, acc);
    }
    const float bias = B2[n];
#pragma unroll
    for (int v = 0; v < 8; v++) {
      float h = acc[v] + bias;
      h = h > 0.f ? h : 0.f;
      h2s[w][(v + 8 * hi) * 64 + n] = h;
    }
  }
  asm volatile("s_wait_dscnt 0x0" ::: "memory");

  // ---- layer 3: (16x64) @ (64x128), K=64 -> 16 wmma steps, 8 col tiles ----
  v8f c3[8];
#pragma unroll
  for (int nt = 0; nt < 8; nt++) {
    const int n = nt * 16 + m;
    v8f acc = {};
#pragma unroll
    for (int kt = 0; kt < 16; kt++) {
      const int k0 = kt * 4 + 2 * hi;
      v2f a;
      a.x = h2s[w][m * 64 + k0];
      a.y = h2s[w][m * 64 + k0 + 1];
      v2f b;
      b.x = W3[k0 * 128 + n];
      b.y = W3[(k0 + 1) * 128 + n];
      acc = wmma4(a, b, acc);
    }
    const float bias = B3[n];
#pragma unroll
    for (int v = 0; v < 8; v++) acc[v] += bias;
    c3[nt] = acc;
  }

  // ---- LayerNorm: per-row stats; rows live in 16-lane halves of C layout ----
  float s[8], ss[8];
#pragma unroll
  for (int v = 0; v < 8; v++) { s[v] = 0.f; ss[v] = 0.f; }
#pragma unroll
  for (int nt = 0; nt < 8; nt++)
#pragma unroll
    for (int v = 0; v < 8; v++) {
      const float h = c3[nt][v];
      s[v] += h;
      ss[v] += h * h;
    }
#pragma unroll
  for (int v = 0; v < 8; v++) {
#pragma unroll
    for (int off = 1; off < 16; off <<= 1) {  // butterflies stay within halves
      s[v] += __shfl_xor(s[v], off, 32);
      ss[v] += __shfl_xor(ss[v], off, 32);
    }
  }
  float mu[8], rstd[8];
#pragma unroll
  for (int v = 0; v < 8; v++) {
    mu[v] = s[v] * (1.0f / 128.0f);
    const float var = ss[v] * (1.0f / 128.0f) - mu[v] * mu[v];
    rstd[v] = rsqrtf(var + 1e-5f);
  }
#pragma unroll
  for (int nt = 0; nt < 8; nt++) {
    const int n = nt * 16 + m;
    const float g = gamma[n], bb = beta[n];
#pragma unroll
    for (int v = 0; v < 8; v++) {
      const float xv = (c3[nt][v] - mu[v]) * rstd[v] * g + bb;
      out_x[(rowTile * 16 + v + 8 * hi) * 128 + n] = xv;
    }
  }
}

// ---------------------------------------------------------------------------
// Pairwise d^2 via WMMA: A row m = [-2x, -2y, |p|^2, 1], B col n = [x, y, 1, |p|^2]
// => C[m][n] = d2(m,n) in one f32 WMMA, full precision.
// ---------------------------------------------------------------------------
__device__ __forceinline__ v2f make_a(const float2* last, int rowTile, int m,
                                      int hi) {
  const float2 r = last[rowTile * 16 + m];  // single b64 load
  v2f a;
  a.x = hi ? (r.x * r.x + r.y * r.y) : (-2.0f * r.x);
  a.y = hi ? 1.0f : (-2.0f * r.y);
  return a;
}

__device__ __forceinline__ v2f make_b(const float2* last, int ct, int m,
                                      int hi) {
  const float2 c = last[ct * 16 + m];  // single b64 load
  v2f b;
  b.x = hi ? 1.0f : c.x;
  b.y = hi ? (c.x * c.x + c.y * c.y) : c.y;
  return b;
}

__global__ void __launch_bounds__(256)
count_kernel(const float2* __restrict__ last, unsigned* __restrict__ row_count) {
  const int lane = threadIdx.x & 31;
  const int w = threadIdx.x >> 5;
  const int m = lane & 15;
  const int hi = lane >> 4;
  const int rowTile = blockIdx.x * 8 + w;
  const float R2 = 0.015f * 0.015f;

  const v2f a = make_a(last, rowTile, m, hi);
  unsigned cnt[8];
#pragma unroll
  for (int v = 0; v < 8; v++) cnt[v] = 0;

  for (int ct = 0; ct < NTILES; ct++) {
    const v2f b = make_b(last, ct, m, hi);
    v8f c = {};
    c = wmma4(a, b, c);
    const bool diag = (ct == rowTile);
#pragma unroll
    for (int v = 0; v < 8; v++) {
      const bool self = diag && (m == v + 8 * hi);
      cnt[v] += ((c[v] <= R2) && !self) ? 1u : 0u;
    }
  }
#pragma unroll
  for (int v = 0; v < 8; v++) {
#pragma unroll
    for (int off = 1; off < 16; off <<= 1)
      cnt[v] += (unsigned)__shfl_xor((int)cnt[v], off, 32);
  }
  if (lane == 0) {
#pragma unroll
    for (int v = 0; v < 8; v++) row_count[rowTile * 16 + v] = cnt[v];
  }
  if (lane == 16) {
#pragma unroll
    for (int v = 0; v < 8; v++) row_count[rowTile * 16 + 8 + v] = cnt[v];
  }
}

__global__ void __launch_bounds__(256)
scan_kernel(const unsigned* __restrict__ rc, unsigned* __restrict__ ro) {
  __shared__ unsigned part[256];
  const int t = threadIdx.x;
  unsigned s = 0;
  for (int i = 0; i < 64; i++) s += rc[t * 64 + i];
  part[t] = s;
  __syncthreads();
  for (int off = 1; off < 256; off <<= 1) {
    const unsigned v = (t >= off) ? part[t - off] : 0u;
    __syncthreads();
    part[t] += v;
    __syncthreads();
  }
  unsigned run = (t == 0) ? 0u : part[t - 1];
  for (int i = 0; i < 64; i++) {
    const unsigned cv = rc[t * 64 + i];
    ro[t * 64 + i] = run;
    run += cv;
  }
  if (t == 255) ro[NPART] = run;  // total edge count
}

__global__ void __launch_bounds__(256)
fill_kernel(const float2* __restrict__ last,
            const unsigned* __restrict__ row_offset, int* __restrict__ srcA,
            int* __restrict__ dstA) {
  const int lane = threadIdx.x & 31;
  const int w = threadIdx.x >> 5;
  const int m = lane & 15;
  const int hi = lane >> 4;
  const int rowTile = blockIdx.x * 8 + w;
  const float R2 = 0.015f * 0.015f;

  const v2f a = make_a(last, rowTile, m, hi);
  unsigned base[8];
#pragma unroll
  for (int v = 0; v < 8; v++) base[v] = row_offset[rowTile * 16 + v + 8 * hi];

  for (int ct = 0; ct < NTILES; ct++) {
    const v2f b = make_b(last, ct, m, hi);
    v8f c = {};
    c = wmma4(a, b, c);
    const bool diag = (ct == rowTile);
#pragma unroll
    for (int v = 0; v < 8; v++) {
      const bool self = diag && (m == v + 8 * hi);
      const bool pred = (c[v] <= R2) && !self;
      const unsigned bal = (unsigned)__ballot(pred);
      const unsigned half = hi ? (bal >> 16) : (bal & 0xFFFFu);
      const unsigned pos = __popc(half & ((1u << m) - 1u));
      if (pred) {
        const unsigned idx = base[v] + pos;
        if (idx < MAXE) {
          srcA[idx] = rowTile * 16 + v + 8 * hi;
          dstA[idx] = ct * 16 + m;
        }
      }
      base[v] += __popc(half);
    }
  }
}

// ---------------------------------------------------------------------------
// edge_attr broadcast (dominant byte stream: 256 MB) + zero-fill padding
// ---------------------------------------------------------------------------
__global__ void __launch_bounds__(256)
attr_kernel(const float* __restrict__ e0, const unsigned* __restrict__ totalp,
            int* __restrict__ srcA, int* __restrict__ dstA,
            float* __restrict__ attr) {
  const unsigned gid = blockIdx.x * 256 + threadIdx.x;
  const unsigned e = gid >> 5;
  const unsigned c4 = gid & 31;
  const unsigned nE = totalp[0];
  const bool valid = e < nE;
  float4 val = make_float4(0.f, 0.f, 0.f, 0.f);
  if (valid) val = reinterpret_cast<const float4*>(e0)[c4];
  reinterpret_cast<float4*>(attr)[e * 32 + c4] = val;
  if (!valid && c4 == 0) {
    srcA[e] = 0;
    dstA[e] = 0;
  }
}

// ---------------------------------------------------------------------------
extern "C" void kernel_launch(void* const* d_in, const int* in_sizes, int n_in,
                              void* d_out, int out_size, void* d_ws,
                              size_t ws_size, hipStream_t stream) {
  const float* pos    = (const float*)d_in[0];
  const float* bounds = (const float*)d_in[1];
  const float* W1     = (const float*)d_in[2];
  const float* B1     = (const float*)d_in[3];
  const float* W2     = (const float*)d_in[4];
  const float* B2     = (const float*)d_in[5];
  const float* W3     = (const float*)d_in[6];
  const float* B3     = (const float*)d_in[7];
  const float* gamma  = (const float*)d_in[8];
  const float* beta   = (const float*)d_in[9];
  const float* e0     = (const float*)d_in[10];

  float* out = (float*)d_out;
  float* out_x = out;                                  // N*128
  int* srcA = (int*)(out + NPART * 128);               // MAXE
  int* dstA = srcA + MAXE;                             // MAXE
  float* attr = out + NPART * 128 + 2 * MAXE;          // MAXE*128

  float2* last = (float2*)d_ws;                                  // N float2
  unsigned* row_count = (unsigned*)((char*)d_ws + 2 * NPART * 4);  // N u32
  unsigned* row_offset = row_count + NPART;                      // N+1 u32

  prep_last_kernel<<<NPART / 256, 256, 0, stream>>>(pos, last);
  mlp_kernel<<<NTILES / 4, 128, 0, stream>>>(pos, bounds, W1, B1, W2, B2, W3,
                                             B3, gamma, beta, out_x);
  count_kernel<<<NTILES / 8, 256, 0, stream>>>(last, row_count);
  scan_kernel<<<1, 256, 0, stream>>>(row_count, row_offset);
  fill_kernel<<<NTILES / 8, 256, 0, stream>>>(last, row_offset, srcA, dstA);
  attr_kernel<<<(MAXE * 32) / 256, 256, 0, stream>>>(e0, row_offset + NPART,
                                                     srcA, dstA, attr);
}